// SpectralConv3d_15006615734082
// MI455X (gfx1250) — compile-verified
//
#include <hip/hip_runtime.h>

typedef __attribute__((ext_vector_type(2))) float v2f;
typedef __attribute__((ext_vector_type(8))) float v8f;

#define PI_F 3.14159265358979323846f

__device__ __forceinline__ v8f wmma4(v2f a, v2f b, v8f c) {
  // D = A(16x4 f32) * B(4x16 f32) + C(16x16 f32)
  return __builtin_amdgcn_wmma_f32_16x16x4_f32(false, a, false, b, (short)0, c,
                                               false, false);
}

// CDNA5 async DMA: global memory -> LDS, 16B per lane, tracked by ASYNCcnt.
__device__ __forceinline__ unsigned lds_off(const void* p) {
  return (unsigned)(unsigned long long)p; // low 32 bits of generic addr = LDS offset
}
__device__ __forceinline__ void async_load_b128(unsigned lds_addr,
                                                const void* gaddr) {
  asm volatile("global_load_async_to_lds_b128 %0, %1, off" ::"v"(lds_addr),
               "v"((unsigned long long)gaddr)
               : "memory");
}
__device__ __forceinline__ void wait_async0() {
  asm volatile("s_wait_asynccnt 0x0" ::: "memory");
}

// ---------------------------------------------------------------------------
// Stage 0: tiled transpose weights (Cin,Cout,16,16,16) -> Wt[mode][i][o]
// Coalesced reads along mode, coalesced writes along o.
// ---------------------------------------------------------------------------
__global__ void k_wt(const float* __restrict__ w, float* __restrict__ Wt) {
  __shared__ float t[64][65];
  int i = blockIdx.x & 63;  // Cin
  int mb = blockIdx.x >> 6; // mode block of 64
  int tid = threadIdx.x;
  for (int idx = tid; idx < 4096; idx += 256) {
    int o = idx >> 6, m = idx & 63;
    t[o][m] = w[((long long)i * 64 + o) * 4096 + mb * 64 + m];
  }
  __syncthreads();
  for (int idx = tid; idx < 4096; idx += 256) {
    int m = idx >> 6, o = idx & 63;
    Wt[(((long long)(mb * 64 + m)) * 64 + i) * 64 + o] = t[o][m];
  }
}

// ---------------------------------------------------------------------------
// Stage 1: rFFT along W, keep 16 modes.  GEMM: (1M x 64) x (64 x 32)
// X1[row][k] : k<16 real part, k>=16 imag part.  row=(b,ci,d,h)
// ---------------------------------------------------------------------------
__global__ void k_fft_w(const float* __restrict__ x, float* __restrict__ X1) {
  __shared__ float basis[64 * 32];    // [n][k]: cos | -sin
  __shared__ float atile[8][16 * 64]; // per-wave 16-row A tile
  int tid = threadIdx.x;
  int wave = tid >> 5, lane = tid & 31, half = lane >> 4, l16 = lane & 15;
  long long r0 = ((long long)blockIdx.x * 8 + wave) * 16;
  { // async DMA the 4KB A tile into LDS (8 x b128 per lane-set)
    unsigned lb = lds_off(&atile[wave][0]);
    const char* gb = (const char*)(x + r0 * 64);
    for (int j = 0; j < 8; ++j) {
      int f = lane + 32 * j;
      async_load_b128(lb + f * 16, gb + f * 16);
    }
  }
  for (int i = tid; i < 64 * 32; i += 256) {
    int n = i >> 5, k = i & 31;
    float th = 2.0f * PI_F * (float)(k & 15) * (float)n * (1.0f / 64.0f);
    basis[i] = (k < 16) ? __cosf(th) : -__sinf(th);
  }
  wait_async0();
  __syncthreads();
  v8f acc0 = {0, 0, 0, 0, 0, 0, 0, 0};
  v8f acc1 = {0, 0, 0, 0, 0, 0, 0, 0};
  for (int kk = 0; kk < 16; ++kk) {
    int k0 = kk * 4 + 2 * half;
    v2f a, b0, b1;
    a[0] = atile[wave][l16 * 64 + k0];
    a[1] = atile[wave][l16 * 64 + k0 + 1];
    b0[0] = basis[k0 * 32 + l16];
    b0[1] = basis[(k0 + 1) * 32 + l16];
    b1[0] = basis[k0 * 32 + 16 + l16];
    b1[1] = basis[(k0 + 1) * 32 + 16 + l16];
    acc0 = wmma4(a, b0, acc0);
    acc1 = wmma4(a, b1, acc1);
  }
  for (int r = 0; r < 8; ++r) {
    long long row = r0 + r + 8 * half;
    X1[row * 32 + l16] = acc0[r];
    X1[row * 32 + 16 + l16] = acc1[r];
  }
}

// ---------------------------------------------------------------------------
// Stage 2: complex DFT along H, keep 16 modes.  Per g=(b,ci,d): M=k3(16),
// K=h(64), N=k2(16).  X2[g][k2][k3][comp]
// ---------------------------------------------------------------------------
__global__ void k_fft_h(const float* __restrict__ X1, float* __restrict__ X2) {
  __shared__ float cb[64 * 16]; // cos(2*pi*k2*h/64) [h][k2]
  __shared__ float sb[64 * 16];
  __shared__ float slice[8][64 * 32]; // per-wave input slice [h][re16|im16]
  int tid = threadIdx.x;
  int wave = tid >> 5, lane = tid & 31, half = lane >> 4, l16 = lane & 15;
  long long g = (long long)blockIdx.x * 8 + wave;
  { // async DMA the 8KB slice into LDS (16 x b128 per lane-set)
    unsigned lb = lds_off(&slice[wave][0]);
    const char* gb = (const char*)(X1 + g * 2048);
    for (int j = 0; j < 16; ++j) {
      int f = lane + 32 * j;
      async_load_b128(lb + f * 16, gb + f * 16);
    }
  }
  for (int i = tid; i < 64 * 16; i += 256) {
    int h = i >> 4, k = i & 15;
    float th = 2.0f * PI_F * (float)k * (float)h * (1.0f / 64.0f);
    cb[i] = __cosf(th);
    sb[i] = __sinf(th);
  }
  wait_async0();
  __syncthreads();
  v8f rc = {0, 0, 0, 0, 0, 0, 0, 0}, rs = rc, ic = rc, is = rc;
  for (int kk = 0; kk < 16; ++kk) {
    int k0 = kk * 4;
    v2f ar, ai, bcf, bsf;
    for (int v = 0; v < 2; ++v) {
      int h = k0 + 2 * half + v;
      ar[v] = slice[wave][h * 32 + l16];
      ai[v] = slice[wave][h * 32 + 16 + l16];
      bcf[v] = cb[h * 16 + l16];
      bsf[v] = sb[h * 16 + l16];
    }
    rc = wmma4(ar, bcf, rc);
    rs = wmma4(ar, bsf, rs);
    ic = wmma4(ai, bcf, ic);
    is = wmma4(ai, bsf, is);
  }
  v8f re = rc + is; // (Xr+iXi)(c-is) real
  v8f im = ic - rs; // imag
  for (int r = 0; r < 8; ++r) {
    int k3 = r + 8 * half, k2 = l16;
    long long o = ((g * 16 + k2) * 16 + k3) * 2;
    X2[o] = re[r];
    X2[o + 1] = im[r];
  }
}

// ---------------------------------------------------------------------------
// Stage 3: complex DFT along D, keep 16 modes.  Per (bc, mtile): M=(k2,k3)
// rows, K=d(64), N=k1(16).  X3[mode][b][comp][ci]
// ---------------------------------------------------------------------------
__global__ void k_fft_d(const float* __restrict__ X2, float* __restrict__ X3) {
  __shared__ float cb[64 * 16]; // [d][k1]
  __shared__ float sb[64 * 16];
  int tid = threadIdx.x;
  for (int i = tid; i < 64 * 16; i += 256) {
    int d = i >> 4, k = i & 15;
    float th = 2.0f * PI_F * (float)k * (float)d * (1.0f / 64.0f);
    cb[i] = __cosf(th);
    sb[i] = __sinf(th);
  }
  __syncthreads();
  int wave = tid >> 5, lane = tid & 31, half = lane >> 4, l16 = lane & 15;
  int id = blockIdx.x * 8 + wave; // 0..4095
  int bc = id >> 4;               // (b*64+ci)
  int mt = id & 15;
  int b = bc >> 6, ci = bc & 63;
  int mrow = mt * 16 + l16; // k2*16+k3
  v8f rc = {0, 0, 0, 0, 0, 0, 0, 0}, rs = rc, ic = rc, is = rc;
  for (int kk = 0; kk < 16; ++kk) {
    int k0 = kk * 4;
    v2f ar, ai, bcf, bsf;
    for (int v = 0; v < 2; ++v) {
      int d = k0 + 2 * half + v;
      long long idx = (((long long)bc * 64 + d) * 256 + mrow) * 2;
      ar[v] = X2[idx];
      ai[v] = X2[idx + 1];
      bcf[v] = cb[d * 16 + l16];
      bsf[v] = sb[d * 16 + l16];
    }
    rc = wmma4(ar, bcf, rc);
    rs = wmma4(ar, bsf, rs);
    ic = wmma4(ai, bcf, ic);
    is = wmma4(ai, bsf, is);
  }
  v8f re = rc + is;
  v8f im = ic - rs;
  for (int r = 0; r < 8; ++r) {
    int m = mt * 16 + r + 8 * half;
    long long mode = (long long)l16 * 256 + m; // k1*256 + k2*16 + k3
    long long o = (mode * 4 + b) * 128 + ci;
    X3[o] = re[r];      // comp 0 (re)
    X3[o + 64] = im[r]; // comp 1 (im)
  }
}

// ---------------------------------------------------------------------------
// Stage 4: per-mode channel mix.  M=(b,comp)=8 rows (padded to 16),
// K=ci(64), N=co(64); real weights, so re/im are just extra rows.
// ---------------------------------------------------------------------------
__global__ void k_mix(const float* __restrict__ X3,
                      const float* __restrict__ Wt, float* __restrict__ Y3) {
  int tid = threadIdx.x;
  int wave = tid >> 5, lane = tid & 31, half = lane >> 4, l16 = lane & 15;
  int mode = blockIdx.x * 8 + wave;
  const float* A = X3 + (long long)mode * 512;   // [m8][ci]
  const float* Bm = Wt + (long long)mode * 4096; // [i][o]
  for (int nt = 0; nt < 4; ++nt) {
    v8f acc = {0, 0, 0, 0, 0, 0, 0, 0};
    for (int kk = 0; kk < 16; ++kk) {
      int k0 = kk * 4;
      v2f a, bf;
      for (int v = 0; v < 2; ++v) {
        int k = k0 + 2 * half + v;
        a[v] = (l16 < 8) ? A[l16 * 64 + k] : 0.0f; // pad rows 8..15
        bf[v] = Bm[k * 64 + nt * 16 + l16];
      }
      acc = wmma4(a, bf, acc);
    }
    if (half == 0) { // rows 0..7 hold valid (b,comp)
      for (int r = 0; r < 8; ++r)
        Y3[(long long)mode * 512 + r * 64 + nt * 16 + l16] = acc[r];
    }
  }
}

// ---------------------------------------------------------------------------
// Stage 5: inverse DFT over D (16 modes -> 64), basis e^{+i th}/64.
// Per (bo, mtile): M=(k2,k3), K=k1(16), N=d(64).  Y2[(bo,d)][k2][k3][comp]
// ---------------------------------------------------------------------------
__global__ void k_ifft_d(const float* __restrict__ Y3, float* __restrict__ Y2) {
  __shared__ float cb[16 * 64]; // [k1][d], scaled by 1/64
  __shared__ float sb[16 * 64];
  int tid = threadIdx.x;
  for (int i = tid; i < 16 * 64; i += 256) {
    int k = i >> 6, d = i & 63;
    float th = 2.0f * PI_F * (float)k * (float)d * (1.0f / 64.0f);
    cb[i] = __cosf(th) * (1.0f / 64.0f);
    sb[i] = __sinf(th) * (1.0f / 64.0f);
  }
  __syncthreads();
  int wave = tid >> 5, lane = tid & 31, half = lane >> 4, l16 = lane & 15;
  int id = blockIdx.x * 8 + wave; // 0..4095
  int bo = id >> 4;               // b*64+o
  int mt = id & 15;
  int b = bo >> 6, o = bo & 63;
  int mrow = mt * 16 + l16;
  for (int nt = 0; nt < 4; ++nt) {
    v8f rc = {0, 0, 0, 0, 0, 0, 0, 0}, rs = rc, ic = rc, is = rc;
    for (int kk = 0; kk < 4; ++kk) {
      int k0 = kk * 4;
      v2f ar, ai, bcf, bsf;
      for (int v = 0; v < 2; ++v) {
        int k1 = k0 + 2 * half + v;
        long long idx = (((long long)k1 * 256 + mrow) * 4 + b) * 128 + o;
        ar[v] = Y3[idx];
        ai[v] = Y3[idx + 64];
        bcf[v] = cb[k1 * 64 + nt * 16 + l16];
        bsf[v] = sb[k1 * 64 + nt * 16 + l16];
      }
      rc = wmma4(ar, bcf, rc);
      rs = wmma4(ar, bsf, rs);
      ic = wmma4(ai, bcf, ic);
      is = wmma4(ai, bsf, is);
    }
    v8f re = rc - is; // (Yr+iYi)(c+is) real
    v8f im = rs + ic; // imag
    for (int r = 0; r < 8; ++r) {
      int m = mt * 16 + r + 8 * half;
      int k2 = m >> 4, k3 = m & 15;
      int d = nt * 16 + l16;
      long long oi = ((((long long)bo * 64 + d) * 16 + k2) * 16 + k3) * 2;
      Y2[oi] = re[r];
      Y2[oi + 1] = im[r];
    }
  }
}

// ---------------------------------------------------------------------------
// Stage 6: inverse DFT over H (16 -> 64).  Per g2=(bo,d): M=k3(16),
// K=k2(16), N=h(64).  Y1[(g2,h)][k3 re16 | k3 im16]
// ---------------------------------------------------------------------------
__global__ void k_ifft_h(const float* __restrict__ Y2, float* __restrict__ Y1) {
  __shared__ float cb[16 * 64]; // [k2][h], scaled by 1/64
  __shared__ float sb[16 * 64];
  int tid = threadIdx.x;
  for (int i = tid; i < 16 * 64; i += 256) {
    int k = i >> 6, h = i & 63;
    float th = 2.0f * PI_F * (float)k * (float)h * (1.0f / 64.0f);
    cb[i] = __cosf(th) * (1.0f / 64.0f);
    sb[i] = __sinf(th) * (1.0f / 64.0f);
  }
  __syncthreads();
  int wave = tid >> 5, lane = tid & 31, half = lane >> 4, l16 = lane & 15;
  long long g2 = (long long)blockIdx.x * 8 + wave;
  for (int nt = 0; nt < 4; ++nt) {
    v8f rc = {0, 0, 0, 0, 0, 0, 0, 0}, rs = rc, ic = rc, is = rc;
    for (int kk = 0; kk < 4; ++kk) {
      int k0 = kk * 4;
      v2f ar, ai, bcf, bsf;
      for (int v = 0; v < 2; ++v) {
        int k2 = k0 + 2 * half + v;
        long long idx = ((g2 * 16 + k2) * 16 + l16) * 2;
        ar[v] = Y2[idx];
        ai[v] = Y2[idx + 1];
        bcf[v] = cb[k2 * 64 + nt * 16 + l16];
        bsf[v] = sb[k2 * 64 + nt * 16 + l16];
      }
      rc = wmma4(ar, bcf, rc);
      rs = wmma4(ar, bsf, rs);
      ic = wmma4(ai, bcf, ic);
      is = wmma4(ai, bsf, is);
    }
    v8f re = rc - is;
    v8f im = rs + ic;
    for (int r = 0; r < 8; ++r) {
      int k3 = r + 8 * half;
      int h = nt * 16 + l16;
      long long oi = (g2 * 64 + h) * 32;
      Y1[oi + k3] = re[r];
      Y1[oi + 16 + k3] = im[r];
    }
  }
}

// ---------------------------------------------------------------------------
// Stage 7: inverse rFFT over W (16 complex modes -> 64 reals) + bias.
// GEMM (1M x 32) x (32 x 64), Hermitian weights folded into the basis.
// ---------------------------------------------------------------------------
__global__ void k_ifft_w(const float* __restrict__ Y1,
                         const float* __restrict__ bias,
                         float* __restrict__ out) {
  __shared__ float basis[32 * 64];    // [k][n]
  __shared__ float atile[8][16 * 32]; // per-wave 16-row tile
  int tid = threadIdx.x;
  int wave = tid >> 5, lane = tid & 31, half = lane >> 4, l16 = lane & 15;
  long long r0 = ((long long)blockIdx.x * 8 + wave) * 16;
  { // async DMA the 2KB A tile into LDS (4 x b128 per lane-set)
    unsigned lb = lds_off(&atile[wave][0]);
    const char* gb = (const char*)(Y1 + r0 * 32);
    for (int j = 0; j < 4; ++j) {
      int f = lane + 32 * j;
      async_load_b128(lb + f * 16, gb + f * 16);
    }
  }
  for (int i = tid; i < 32 * 64; i += 256) {
    int k = i >> 6, n = i & 63;
    int km = k & 15;
    float th = 2.0f * PI_F * (float)km * (float)n * (1.0f / 64.0f);
    float wgt = ((km == 0) ? 1.0f : 2.0f) * (1.0f / 64.0f);
    basis[i] = ((k < 16) ? __cosf(th) : -__sinf(th)) * wgt;
  }
  wait_async0();
  __syncthreads();
  int o = (int)((r0 >> 12) & 63); // row = ((b*64+o)*64+d)*64+h
  float bval = bias[o];
  for (int nt = 0; nt < 4; ++nt) {
    v8f acc = {0, 0, 0, 0, 0, 0, 0, 0};
    for (int kk = 0; kk < 8; ++kk) {
      int k0 = kk * 4 + 2 * half;
      v2f a, bf;
      a[0] = atile[wave][l16 * 32 + k0];
      a[1] = atile[wave][l16 * 32 + k0 + 1];
      bf[0] = basis[k0 * 64 + nt * 16 + l16];
      bf[1] = basis[(k0 + 1) * 64 + nt * 16 + l16];
      acc = wmma4(a, bf, acc);
    }
    for (int r = 0; r < 8; ++r) {
      long long row = r0 + r + 8 * half;
      out[row * 64 + nt * 16 + l16] = acc[r] + bval;
    }
  }
}

// ---------------------------------------------------------------------------
extern "C" void kernel_launch(void* const* d_in, const int* in_sizes, int n_in,
                              void* d_out, int out_size, void* d_ws,
                              size_t ws_size, hipStream_t stream) {
  (void)in_sizes;
  (void)n_in;
  (void)out_size;
  (void)ws_size;
  const float* x = (const float*)d_in[0];
  const float* w = (const float*)d_in[1];
  const float* bias = (const float*)d_in[2];
  float* out = (float*)d_out;
  float* ws = (float*)d_ws;

  // Workspace layout (floats). Y1 aliases X1, Y2 aliases X2 (dead by then).
  float* X1 = ws;              // 33,554,432
  float* X2 = X1 + 33554432LL; //  8,388,608
  float* X3 = X2 + 8388608LL;  //  2,097,152
  float* Y3 = X3 + 2097152LL;  //  2,097,152
  float* Wt = Y3 + 2097152LL;  // 16,777,216  (total ~252 MB)
  float* Y2 = X2;
  float* Y1 = X1;

  k_wt<<<4096, 256, 0, stream>>>(w, Wt);
  k_fft_w<<<8192, 256, 0, stream>>>(x, X1);
  k_fft_h<<<2048, 256, 0, stream>>>(X1, X2);
  k_fft_d<<<512, 256, 0, stream>>>(X2, X3);
  k_mix<<<512, 256, 0, stream>>>(X3, Wt, Y3);
  k_ifft_d<<<512, 256, 0, stream>>>(Y3, Y2);
  k_ifft_h<<<2048, 256, 0, stream>>>(Y2, Y1);
  k_ifft_w<<<8192, 256, 0, stream>>>(Y1, bias, out);
}